// AudioTokenFilter_83004537962991
// MI455X (gfx1250) — compile-verified
//
#include <hip/hip_runtime.h>
#include <hip/hip_bf16.h>

// Problem constants (from reference setup_inputs / hyperparameters)
#define B_ 16
#define L_ 4096
#define D_ 1024
#define C_ 512
#define K_ 2048   // min(max(ceil(L*0.5), 10), L)

// ---------------------------------------------------------------------------
// Phase 1: per-token negative-entropy score s = sum_c p*log(p+1e-8).
// Ordering by s == ordering by rel = 1 - H/logC (strictly monotone), so we
// rank on s directly. One wave32 per token: 4 coalesced float4 loads per lane
// (32 lanes * 16 floats = 512 classes), then a 5-step shuffle tree.
// ---------------------------------------------------------------------------
__global__ __launch_bounds__(256) void AudioTokenFilter_rel_kernel(
    const float* __restrict__ post, float* __restrict__ rel) {
  const int wave = threadIdx.x >> 5;
  const int lane = threadIdx.x & 31;
  const int token = blockIdx.x * 8 + wave;
  const float4* __restrict__ p4 = (const float4*)(post + (size_t)token * C_);
  float s = 0.0f;
#pragma unroll
  for (int c = 0; c < 4; ++c) {
    float4 v = p4[c * 32 + lane];
    s += v.x * __builtin_logf(v.x + 1e-8f);
    s += v.y * __builtin_logf(v.y + 1e-8f);
    s += v.z * __builtin_logf(v.z + 1e-8f);
    s += v.w * __builtin_logf(v.w + 1e-8f);
  }
#pragma unroll
  for (int off = 16; off > 0; off >>= 1) s += __shfl_xor(s, off, 32);
  if (lane == 0) rel[token] = s;  // higher s  <=>  higher reliability
}

// ---------------------------------------------------------------------------
// Phase 2: per-row exact top-K with jax.lax.top_k tie semantics, output
// positions directly in ascending token order (== jnp.sort(top_idx)).
// One 1024-thread workgroup per batch row; keys in LDS; 4-pass MSB radix
// select for the K-th largest key; packed (gt,eq) exclusive scan -> compaction.
// ---------------------------------------------------------------------------
__device__ __forceinline__ unsigned f2key(float f) {
  unsigned u = __float_as_uint(f);
  return (u & 0x80000000u) ? ~u : (u | 0x80000000u);  // larger float -> larger key
}

__global__ __launch_bounds__(1024) void AudioTokenFilter_select_kernel(
    const float* __restrict__ rel, float* __restrict__ outIdx,
    float* __restrict__ outMask) {
  __shared__ unsigned keys[L_];      // 16 KB
  __shared__ unsigned hist[256];
  __shared__ unsigned scanb[1024];
  __shared__ unsigned sDigit;
  __shared__ int sR;
  __shared__ unsigned sG;

  const int b = blockIdx.x;
  const int t = threadIdx.x;

  // Load row (coalesced float4) and convert to sortable keys.
  const float4 r = ((const float4*)(rel + (size_t)b * L_))[t];
  keys[4 * t + 0] = f2key(r.x);
  keys[4 * t + 1] = f2key(r.y);
  keys[4 * t + 2] = f2key(r.z);
  keys[4 * t + 3] = f2key(r.w);
  if (t == 0) { sR = K_; sG = 0u; }
  __syncthreads();

  // 4-pass MSB radix select: find K-th largest key T and G = #{key > T}.
  unsigned prefix = 0u;
#pragma unroll
  for (int pass = 0; pass < 4; ++pass) {
    const int shift = 24 - 8 * pass;
    if (t < 256) hist[t] = 0u;
    __syncthreads();
#pragma unroll
    for (int j = 0; j < 4; ++j) {
      unsigned key = keys[4 * t + j];
      bool cand = (pass == 0) || ((key >> (shift + 8)) == prefix);
      if (cand) atomicAdd(&hist[(key >> shift) & 255u], 1u);
    }
    __syncthreads();
    if (t == 0) {
      unsigned c = 0u;
      for (int bin = 255; bin >= 0; --bin) {
        unsigned h = hist[bin];
        if (c + h >= (unsigned)sR) { sDigit = (unsigned)bin; sG += c; sR -= (int)c; break; }
        c += h;
      }
    }
    __syncthreads();
    prefix = (prefix << 8) | sDigit;
  }
  const unsigned T = prefix;            // K-th largest key
  const unsigned G = sG;                // strictly-greater count
  const unsigned E = (unsigned)K_ - G;  // ties at T to keep (lowest indices), E >= 1

  // Packed exclusive scan in index order: high 16 = #greater, low 16 = #equal.
  unsigned myKey[4], pkEx[4], run = 0u;
#pragma unroll
  for (int j = 0; j < 4; ++j) {
    unsigned key = keys[4 * t + j];
    myKey[j] = key;
    unsigned pk = ((key > T) ? 0x10000u : 0u) | ((key == T) ? 1u : 0u);
    pkEx[j] = run;
    run += pk;
  }
  scanb[t] = run;
  __syncthreads();
  for (int off = 1; off < 1024; off <<= 1) {
    unsigned v = (t >= off) ? scanb[t - off] : 0u;
    __syncthreads();
    scanb[t] += v;
    __syncthreads();
  }
  const unsigned excl = scanb[t] - run;  // exclusive prefix for this thread

#pragma unroll
  for (int j = 0; j < 4; ++j) {
    unsigned tot = excl + pkEx[j];
    unsigned gtEx = tot >> 16, eqEx = tot & 0xFFFFu;
    unsigned key = myKey[j];
    bool sel = (key > T) || (key == T && eqEx < E);
    if (sel) {
      unsigned pos = gtEx + (eqEx < E ? eqEx : E);  // unique, ascending in token idx
      outIdx[(size_t)b * K_ + pos] = (float)(4 * t + j);
    }
  }
  for (int k = t; k < K_; k += 1024) outMask[(size_t)b * K_ + k] = 1.0f;
}

// ---------------------------------------------------------------------------
// Phase 3: gather selected hidden rows (4 KB each). Uses the CDNA5 async
// global<->LDS DMA path (ASYNCcnt) when the builtins are available; otherwise
// a plain 128-bit VGPR copy. Either way the access is fully coalesced b128.
// ---------------------------------------------------------------------------
#if defined(__HIP_DEVICE_COMPILE__) &&                                   \
    __has_builtin(__builtin_amdgcn_global_load_async_to_lds_b128) &&     \
    __has_builtin(__builtin_amdgcn_global_store_async_from_lds_b128) &&  \
    __has_builtin(__builtin_amdgcn_s_wait_asynccnt)
#define ATF_USE_ASYNC_LDS 1
#endif

#ifdef ATF_USE_ASYNC_LDS
// Builtins take pointers to int __attribute__((vector_size(16))) in AS1/AS3
// (per the exact hipcc diagnostic from the previous round).
typedef int v4i_atf __attribute__((vector_size(16)));
typedef __attribute__((address_space(1))) v4i_atf* g_v4i_atf;
typedef __attribute__((address_space(3))) v4i_atf* l_v4i_atf;
#endif

__global__ __launch_bounds__(256) void AudioTokenFilter_gather_kernel(
    const float* __restrict__ hidden, const float* __restrict__ idxf,
    float* __restrict__ outHid) {
  const int row = blockIdx.x;            // 0 .. B*K-1
  const int b = row >> 11;               // row / K_
  const int token = (int)idxf[row];      // exact small integer stored as float
  const float4* __restrict__ src =
      (const float4*)(hidden + ((size_t)b * L_ + (size_t)token) * D_);
  float4* __restrict__ dst = (float4*)(outHid + (size_t)row * D_);
  const int t = threadIdx.x;             // 256 threads x float4 = 4 KB row

#ifdef ATF_USE_ASYNC_LDS
  __shared__ float4 buf[256];
  v4i_atf* gsrc = (v4i_atf*)(void*)(const_cast<float4*>(src) + t);
  v4i_atf* gdst = (v4i_atf*)(void*)(dst + t);
  v4i_atf* lbuf = (v4i_atf*)(void*)&buf[t];
  // global -> LDS (async DMA, no VGPR round-trip)
  __builtin_amdgcn_global_load_async_to_lds_b128(
      (g_v4i_atf)gsrc, (l_v4i_atf)lbuf, 0, 0);
  __builtin_amdgcn_s_wait_asynccnt(0);   // own wave's load done; same-wave LDS reuse
  // LDS -> global (async DMA)
  __builtin_amdgcn_global_store_async_from_lds_b128(
      (g_v4i_atf)gdst, (l_v4i_atf)lbuf, 0, 0);
  __builtin_amdgcn_s_wait_asynccnt(0);
#else
  dst[t] = src[t];
#endif
}

// ---------------------------------------------------------------------------
// Launch: out = [hidden B*K*D | idx B*K | mask B*K] (all f32).
// ---------------------------------------------------------------------------
extern "C" void kernel_launch(void* const* d_in, const int* in_sizes, int n_in,
                              void* d_out, int out_size, void* d_ws, size_t ws_size,
                              hipStream_t stream) {
  const float* hidden = (const float*)d_in[0];
  const float* post = (const float*)d_in[1];

  float* out = (float*)d_out;
  float* outHid = out;                                   // B*K*D
  float* outIdx = out + (size_t)B_ * K_ * D_;            // B*K
  float* outMask = outIdx + (size_t)B_ * K_;             // B*K

  // Reliability scratch (B*L floats = 256 KB). Prefer d_ws; if the workspace
  // is too small, stage it at the front of the hidden output region: it is
  // consumed by the select kernel and only afterwards overwritten by gather.
  float* rel = (ws_size >= (size_t)B_ * L_ * sizeof(float)) ? (float*)d_ws
                                                            : outHid;

  AudioTokenFilter_rel_kernel<<<(B_ * L_) / 8, 256, 0, stream>>>(post, rel);
  AudioTokenFilter_select_kernel<<<B_, 1024, 0, stream>>>(rel, outIdx, outMask);
  AudioTokenFilter_gather_kernel<<<B_ * K_, 256, 0, stream>>>(hidden, outIdx, outHid);
}